// AttentionManifold_43018392437200
// MI455X (gfx1250) — compile-verified
//
#include <hip/hip_runtime.h>
#include <math.h>

// ---------------------------------------------------------------------------
// AttentionManifold on MI455X (gfx1250, wave32, WMMA f32 16x16x4)
//
// Algebraic reduction (see analysis): with A_j = Q_j Q_j^T, B_i = K_i K_i^T
// rank-16 projectors, energy[b,i,j] = ||(A_j-B_i)^2||_F
//                                   = sqrt(2*(16 - 2*||C||_F^2 + ||C C^T||_F^2)),
// C = Q_j^T K_i (16x16). Only V at m = M-1 contributes to the output.
// Cholesky-QR gives the same projectors as jnp.linalg.qr.
// ---------------------------------------------------------------------------

typedef __attribute__((ext_vector_type(2))) float v2f;
typedef __attribute__((ext_vector_type(8))) float v8f;

#define BSZ   4
#define MM_   64
#define DIN   128
#define DOUT  64
#define PP    16

// workspace layout (in floats)
#define OFF_Q   0
#define OFF_K   (BSZ*MM_*DOUT*PP)            // 262144
#define OFF_VP  (2*BSZ*MM_*DOUT*PP)          // 524288
#define OFF_S   (OFF_VP + BSZ*DOUT*DOUT)     // 540672
#define OFF_W   (OFF_S  + BSZ*MM_*MM_)       // 557056

__device__ __forceinline__ v8f wmma_f32(v2f a, v2f b, v8f c) {
  // v_wmma_f32_16x16x4_f32 : D = A(16x4) * B(4x16) + C(16x16)
  return __builtin_amdgcn_wmma_f32_16x16x4_f32(
      /*neg_a=*/false, a, /*neg_b=*/false, b,
      /*c_mod=*/(short)0, c, /*reuse_a=*/false, /*reuse_b=*/false);
}

// ---------------------------------------------------------------------------
// Kernel A: per (b, m, head): Y = W @ x[b,m]  (64x128 @ 128x16, WMMA),
// then Cholesky-QR: G = Y^T Y (WMMA), G = L L^T, Q = Y L^-T (fwd subst).
// head 0 -> Q frame, head 1 -> K frame, head 2 (only m == M-1) -> Vp = Q Q^T.
// One wave (32 threads) per block.
// ---------------------------------------------------------------------------
__global__ __launch_bounds__(32)
void frmap_cholqr_kernel(const float* __restrict__ x,
                         const float* __restrict__ Wq,
                         const float* __restrict__ Wk,
                         const float* __restrict__ Wv,
                         float* __restrict__ qf,
                         float* __restrict__ kf,
                         float* __restrict__ vp)
{
  __shared__ __align__(16) float sW[DOUT*DIN];   // 32 KB
  __shared__ __align__(16) float sX[DIN*PP];     // 8 KB
  __shared__ __align__(16) float sY[DOUT*PP];    // 4 KB
  __shared__ __align__(16) float sG[PP*PP];
  __shared__ __align__(16) float sL[PP*PP];

  const int head = blockIdx.x % 3;
  const int bm   = blockIdx.x / 3;
  const int b    = bm / MM_;
  const int m    = bm % MM_;
  if (head == 2 && m != MM_-1) return;   // only last V frame survives
  const int lane = threadIdx.x;
  const int lc   = lane & 15;            // column / M-row within half-wave
  const int khi  = (lane >> 4) << 1;     // K offset: 0 (lanes 0-15) or 2
  const int rbase= (lane >> 4) << 3;     // C/D: M = r (+8 for upper lanes)

  const float* W   = (head == 0) ? Wq : (head == 1) ? Wk : Wv;
  const float* xbm = x + (size_t)bm * DIN * PP;

  // stage W (64x128) and x[b,m] (128x16, already B-major: rows = K) into LDS
  for (int t = lane; t < DOUT*DIN/4; t += 32)
    ((float4*)sW)[t] = ((const float4*)W)[t];
  for (int t = lane; t < DIN*PP/4; t += 32)
    ((float4*)sX)[t] = ((const float4*)xbm)[t];
  __syncthreads();

  // ---- Y = W @ x : accumulate 32 k-steps of 16x16x4, 4 row-tiles ----------
  v8f acc[4] = {v8f{}, v8f{}, v8f{}, v8f{}};
  for (int kk = 0; kk < DIN/4; ++kk) {
    const int k0 = 4*kk + khi;
    v2f bf = { sX[k0*PP + lc], sX[(k0+1)*PP + lc] };
    #pragma unroll
    for (int t = 0; t < 4; ++t) {
      const int row = t*16 + lc;
      v2f af = { sW[row*DIN + k0], sW[row*DIN + k0 + 1] };
      acc[t] = wmma_f32(af, bf, acc[t]);
    }
  }
  #pragma unroll
  for (int t = 0; t < 4; ++t)
    #pragma unroll
    for (int r = 0; r < 8; ++r)
      sY[(t*16 + r + rbase)*PP + lc] = acc[t][r];
  __syncthreads();

  // ---- Gram G = Y^T Y (16x16, K=64): A-frag == B-frag ---------------------
  v8f g = {};
  for (int kk = 0; kk < DOUT/4; ++kk) {
    const int k0 = 4*kk + khi;
    v2f f = { sY[k0*PP + lc], sY[(k0+1)*PP + lc] };
    g = wmma_f32(f, f, g);
  }
  #pragma unroll
  for (int r = 0; r < 8; ++r) sG[(r + rbase)*PP + lc] = g[r];
  for (int t = lane; t < PP*PP; t += 32) sL[t] = 0.0f;
  __syncthreads();

  // ---- Cholesky G = L L^T (16x16, lanes 0..15 own rows) -------------------
  for (int jj = 0; jj < PP; ++jj) {
    if (lane == 0) {
      float s = sG[jj*PP + jj];
      for (int k = 0; k < jj; ++k) s -= sL[jj*PP + k] * sL[jj*PP + k];
      sL[jj*PP + jj] = sqrtf(fmaxf(s, 1e-20f));
    }
    __syncthreads();
    if (lane > jj && lane < PP) {
      float s = sG[lane*PP + jj];
      for (int k = 0; k < jj; ++k) s -= sL[lane*PP + k] * sL[jj*PP + k];
      sL[lane*PP + jj] = s / sL[jj*PP + jj];
    }
    __syncthreads();
  }

  // ---- Q = Y L^-T : forward substitution, 2 rows per lane -----------------
  float* dstQ = (head == 0) ? (qf + (size_t)bm*DOUT*PP)
              : (head == 1) ? (kf + (size_t)bm*DOUT*PP) : nullptr;
  for (int rr = lane; rr < DOUT; rr += 32) {
    float qrow[PP];
    #pragma unroll
    for (int jj = 0; jj < PP; ++jj) {
      float s = sY[rr*PP + jj];
      for (int k = 0; k < jj; ++k) s -= qrow[k] * sL[jj*PP + k];
      qrow[jj] = s / sL[jj*PP + jj];
    }
    if (head < 2) {
      #pragma unroll
      for (int jj = 0; jj < PP; ++jj) dstQ[rr*PP + jj] = qrow[jj];
    } else {
      #pragma unroll
      for (int jj = 0; jj < PP; ++jj) sY[rr*PP + jj] = qrow[jj]; // own row only
    }
  }

  // ---- head 2: Vp = Q Q^T (64x64, K=16) via WMMA --------------------------
  if (head == 2) {
    __syncthreads();
    float* vp_b = vp + (size_t)b * DOUT * DOUT;
    for (int tm = 0; tm < 4; ++tm) {
      for (int tn = 0; tn < 4; ++tn) {
        v8f p = {};
        #pragma unroll
        for (int kk = 0; kk < 4; ++kk) {
          const int k0 = 4*kk + khi;
          v2f af = { sY[(tm*16 + lc)*PP + k0], sY[(tm*16 + lc)*PP + k0 + 1] };
          v2f bf = { sY[(tn*16 + lc)*PP + k0], sY[(tn*16 + lc)*PP + k0 + 1] };
          p = wmma_f32(af, bf, p);
        }
        #pragma unroll
        for (int r = 0; r < 8; ++r)
          vp_b[(tm*16 + r + rbase)*DOUT + (tn*16 + lc)] = p[r];
      }
    }
  }
}

// ---------------------------------------------------------------------------
// Kernel B: one wave per (b,i,j): C = Q_j^T K_i (WMMA, K=64), D = C C^T
// (WMMA, K=16), score = 1/(1 + log1p(sqrt(2*(16 - 2*||C||^2 + ||D||^2)))).
// ---------------------------------------------------------------------------
__global__ __launch_bounds__(32)
void energy_kernel(const float* __restrict__ qf,
                   const float* __restrict__ kf,
                   float* __restrict__ scores)
{
  __shared__ __align__(16) float sC[PP*PP];
  const int j = blockIdx.x % MM_;
  const int i = (blockIdx.x / MM_) % MM_;
  const int b = blockIdx.x / (MM_*MM_);
  const float* Qj = qf + (size_t)(b*MM_ + j) * DOUT * PP;
  const float* Ki = kf + (size_t)(b*MM_ + i) * DOUT * PP;
  const int lane = threadIdx.x;
  const int lc   = lane & 15;
  const int khi  = (lane >> 4) << 1;
  const int rbase= (lane >> 4) << 3;

  v8f c = {};
  for (int kk = 0; kk < DOUT/4; ++kk) {
    const int k0 = 4*kk + khi;
    v2f af = { Qj[k0*PP + lc], Qj[(k0+1)*PP + lc] };  // A = Q_j^T
    v2f bf = { Ki[k0*PP + lc], Ki[(k0+1)*PP + lc] };  // B = K_i
    c = wmma_f32(af, bf, c);
  }
  float t2 = 0.0f;
  #pragma unroll
  for (int r = 0; r < 8; ++r) {
    t2 += c[r]*c[r];
    sC[(r + rbase)*PP + lc] = c[r];
  }
  __syncthreads();

  v8f d = {};
  #pragma unroll
  for (int kk = 0; kk < 4; ++kk) {
    const int k0 = 4*kk + khi;
    v2f f = { sC[lc*PP + k0], sC[lc*PP + k0 + 1] };   // A = C, B = C^T: same frag
    d = wmma_f32(f, f, d);
  }
  float t4 = 0.0f;
  #pragma unroll
  for (int r = 0; r < 8; ++r) t4 += d[r]*d[r];

  for (int off = 16; off > 0; off >>= 1) {
    t2 += __shfl_xor(t2, off, 32);
    t4 += __shfl_xor(t4, off, 32);
  }
  if (lane == 0) {
    const float tr4 = 2.0f * ((float)PP - 2.0f*t2 + t4);  // tr((A-B)^4)
    const float energy = sqrtf(fmaxf(tr4, 0.0f));
    scores[blockIdx.x] = 1.0f / (1.0f + log1pf(energy));
  }
}

// ---------------------------------------------------------------------------
// Kernel C: w[b,j] = softmax_i(scores[b,i,j])[M-1]
// ---------------------------------------------------------------------------
__global__ __launch_bounds__(MM_)
void softmax_w_kernel(const float* __restrict__ scores, float* __restrict__ w)
{
  const int b = blockIdx.x;
  const int j = threadIdx.x;
  const float* s = scores + (size_t)b*MM_*MM_ + j;
  float mx = -1e30f;
  for (int i = 0; i < MM_; ++i) mx = fmaxf(mx, s[i*MM_]);
  float sum = 0.0f, last = 0.0f;
  for (int i = 0; i < MM_; ++i) {
    const float e = __expf(s[i*MM_] - mx);
    sum += e;
    if (i == MM_-1) last = e;
  }
  w[b*MM_ + j] = last / sum;
}

// ---------------------------------------------------------------------------
// Kernel D: out[b,j,r,c] = w[b,j] * Vp[b,r,c]   (1M fp32, bandwidth-trivial)
// ---------------------------------------------------------------------------
__global__ __launch_bounds__(256)
void out_kernel(const float* __restrict__ w,
                const float* __restrict__ vp,
                float* __restrict__ out)
{
  const int idx = blockIdx.x * 256 + threadIdx.x;
  const int b  = idx >> 18;            // / (M * 64*64)
  const int j  = (idx >> 12) & (MM_-1);
  const int rc = idx & 4095;
  out[idx] = w[b*MM_ + j] * vp[b*4096 + rc];
}

extern "C" void kernel_launch(void* const* d_in, const int* in_sizes, int n_in,
                              void* d_out, int out_size, void* d_ws, size_t ws_size,
                              hipStream_t stream)
{
  (void)in_sizes; (void)n_in; (void)out_size; (void)ws_size;
  const float* x  = (const float*)d_in[0];
  const float* Wq = (const float*)d_in[1];
  const float* Wk = (const float*)d_in[2];
  const float* Wv = (const float*)d_in[3];

  float* ws = (float*)d_ws;
  float* Qf = ws + OFF_Q;   // (BS*M) x 64 x 16 orthonormal frames
  float* Kf = ws + OFF_K;
  float* Vp = ws + OFF_VP;  // BS x 64 x 64 projector of last V frame
  float* Sc = ws + OFF_S;   // BS x M x M scores
  float* Wt = ws + OFF_W;   // BS x M softmax weights

  frmap_cholqr_kernel<<<BSZ*MM_*3, 32, 0, stream>>>(x, Wq, Wk, Wv, Qf, Kf, Vp);
  energy_kernel<<<BSZ*MM_*MM_, 32, 0, stream>>>(Qf, Kf, Sc);
  softmax_w_kernel<<<BSZ, MM_, 0, stream>>>(Sc, Wt);
  out_kernel<<<(BSZ*MM_*DOUT*DOUT)/256, 256, 0, stream>>>(Wt, Vp, (float*)d_out);
}